// VectorQuantizer_62216896250291
// MI455X (gfx1250) — compile-verified
//
#include <hip/hip_runtime.h>
#include <hip/hip_bf16.h>

typedef __attribute__((ext_vector_type(2))) float v2f;
typedef __attribute__((ext_vector_type(8))) float v8f;

constexpr int EMB_D   = 256;     // embedding dim
constexpr int NUM_K   = 4096;    // number of codes
constexpr int NROWS   = 32768;   // 32*32*32
constexpr int M_TILE  = 64;      // rows per workgroup in argmin kernel
constexpr int WAVES   = 8;       // waves per workgroup (wave32)
constexpr int K_PER_WAVE  = NUM_K / WAVES;   // 512
constexpr int NT_PER_WAVE = K_PER_WAVE / 16; // 32 column tiles of 16

// ---------------------------------------------------------------------------
// 1) Transpose V [D,K] -> VT [K,D] so the gather reads contiguous rows.
// ---------------------------------------------------------------------------
__global__ void k_transpose(const float* __restrict__ V, float* __restrict__ VT) {
    __shared__ float tile[32][33];
    const int tk = blockIdx.x * 32;  // k base
    const int td = blockIdx.y * 32;  // d base
    const int x = threadIdx.x;       // 0..31
    const int y = threadIdx.y;       // 0..7
#pragma unroll
    for (int j = 0; j < 32; j += 8)
        tile[y + j][x] = V[(size_t)(td + y + j) * NUM_K + tk + x]; // [d_local][k_local]
    __syncthreads();
#pragma unroll
    for (int j = 0; j < 32; j += 8)
        VT[(size_t)(tk + y + j) * EMB_D + td + x] = tile[x][y + j];
}

// ---------------------------------------------------------------------------
// 2) vnorm[k] = sum_d V[d,k]^2   (coalesced across k)
// ---------------------------------------------------------------------------
__global__ void k_vnorm(const float* __restrict__ V, float* __restrict__ vnorm) {
    const int k = blockIdx.x * blockDim.x + threadIdx.x;
    float s = 0.f;
    for (int d = 0; d < EMB_D; ++d) {
        float v = V[(size_t)d * NUM_K + k];
        s += v * v;
    }
    vnorm[k] = s;
}

// ---------------------------------------------------------------------------
// 3) WMMA argmin: per row, argmin_k (vnorm[k] - 2 * x.v_k)
//    64 rows/WG, 8 waves each own 512 codes. A tile in LDS (XOR-swizzled).
// ---------------------------------------------------------------------------
__global__ __launch_bounds__(256, 1)
void k_argmin(const float* __restrict__ X, const float* __restrict__ V,
              const float* __restrict__ vnorm, int* __restrict__ out_idx,
              float* __restrict__ out_idx_f) {
    __shared__ float ldsA[M_TILE * EMB_D];   // 64 KB, reused as reduce scratch later
    const int t    = threadIdx.x;
    const int wave = t >> 5;
    const int lane = t & 31;
    const int col  = lane & 15;   // B column / A row within subtile
    const int hi   = lane >> 4;   // 0: lanes 0-15, 1: lanes 16-31
    const int sel  = hi * 2;      // K-offset within 4-wide fragment
    const int rowbase = blockIdx.x * M_TILE;

    // Cooperative load of the 64x256 x-tile, swizzled at float4 granularity:
    // physical group = (c4 ^ (row & 15)) -> conflict-free b64 fragment reads.
    {
        const float4* xg = (const float4*)(X + (size_t)rowbase * EMB_D);
#pragma unroll
        for (int i = 0; i < 16; ++i) {
            int idx = t + i * 256;        // float4 index within 64x64 grid
            int r   = idx >> 6;
            int c4  = idx & 63;
            float4 v = xg[idx];
            int g = c4 ^ (r & 15);
            *(float4*)(&ldsA[(r << 8) + (g << 2)]) = v;
        }
    }
    __syncthreads();

    float bestv[4][8];
    int   besti[4][8];
#pragma unroll
    for (int m = 0; m < 4; ++m)
#pragma unroll
        for (int r = 0; r < 8; ++r) { bestv[m][r] = 3.402823466e38f; besti[m][r] = 0; }

    const int k0 = wave * K_PER_WAVE;
    const v8f vzero = {0.f, 0.f, 0.f, 0.f, 0.f, 0.f, 0.f, 0.f};

    for (int nt = 0; nt < NT_PER_WAVE; ++nt) {
        const int kbase = k0 + nt * 16;
        const float* bp = V + (size_t)kbase + col;
        v8f acc[4];
#pragma unroll
        for (int m = 0; m < 4; ++m) acc[m] = vzero;

#pragma unroll 4
        for (int d = 0; d < EMB_D; d += 4) {
            // B fragment: B[d+sel][col], B[d+sel+1][col]  (rows of V, stride K)
            v2f bf;
            bf.x = bp[(size_t)(d + sel) * NUM_K];
            bf.y = bp[(size_t)(d + sel + 1) * NUM_K];
            const int doff = d + sel;
            const int off  = (((doff >> 2) ^ col) << 2) + (doff & 3);
#pragma unroll
            for (int m = 0; m < 4; ++m) {
                const int arow = m * 16 + col;
                v2f af = *(const v2f*)(&ldsA[(arow << 8) + off]);
                acc[m] = __builtin_amdgcn_wmma_f32_16x16x4_f32(
                    false, af, false, bf, (short)0, acc[m], false, false);
            }
        }
        const float vn = vnorm[kbase + col];
        const int   kc = kbase + col;
#pragma unroll
        for (int m = 0; m < 4; ++m)
#pragma unroll
            for (int r = 0; r < 8; ++r) {
                float dist = __builtin_fmaf(-2.f, acc[m][r], vn);
                if (dist < bestv[m][r]) { bestv[m][r] = dist; besti[m][r] = kc; }
            }
    }

    // All waves done reading the A tile -> reuse ldsA as reduce scratch.
    __syncthreads();
    float* scratchV = ldsA;                // [64][8]
    int*   scratchI = (int*)(ldsA + 512);  // [64][8]

#pragma unroll
    for (int m = 0; m < 4; ++m)
#pragma unroll
        for (int r = 0; r < 8; ++r) {
            float v = bestv[m][r];
            int   i = besti[m][r];
#pragma unroll
            for (int mask = 1; mask <= 8; mask <<= 1) {
                float vo = __shfl_xor(v, mask, 32);
                int   io = __shfl_xor(i, mask, 32);
                if (vo < v || (vo == v && io < i)) { v = vo; i = io; }
            }
            if (col == 0) {  // lanes 0 and 16 hold the half-wave minima
                int row = m * 16 + r + hi * 8;
                scratchV[row * 8 + wave] = v;
                scratchI[row * 8 + wave] = i;
            }
        }
    __syncthreads();

    if (t < M_TILE) {
        float v = scratchV[t * 8];
        int   i = scratchI[t * 8];
#pragma unroll
        for (int w = 1; w < WAVES; ++w) {
            float vo = scratchV[t * 8 + w];
            int   io = scratchI[t * 8 + w];
            if (vo < v || (vo == v && io < i)) { v = vo; i = io; }
        }
        out_idx[rowbase + t]   = i;
        out_idx_f[rowbase + t] = (float)i;
    }
}

// ---------------------------------------------------------------------------
// 4) Gather quantized rows from VT, write output, per-block partial MSE sums.
//    8 rows per block of 256 threads (32 threads x 2 float4 per row).
// ---------------------------------------------------------------------------
__global__ void k_gather(const float* __restrict__ X, const float* __restrict__ VT,
                         const int* __restrict__ idx, float* __restrict__ outQ,
                         float* __restrict__ partial) {
    __shared__ float red[256];
    const int t  = threadIdx.x;
    const int rl = t >> 5;
    const int ln = t & 31;
    const int row = blockIdx.x * 8 + rl;
    const int k   = idx[row];
    const float4* q4 = (const float4*)(VT + (size_t)k * EMB_D);
    const float4* x4 = (const float4*)(X + (size_t)row * EMB_D);
    float4* o4 = (float4*)(outQ + (size_t)row * EMB_D);
    float s = 0.f;
#pragma unroll
    for (int j = 0; j < 2; ++j) {
        int c = ln + j * 32;
        float4 q = q4[c];
        float4 x = x4[c];
        o4[c] = q;  // straight-through value == quantized
        float dx = x.x - q.x, dy = x.y - q.y, dz = x.z - q.z, dw = x.w - q.w;
        s += dx * dx + dy * dy + dz * dz + dw * dw;
    }
    red[t] = s;
    __syncthreads();
    for (int off = 128; off > 0; off >>= 1) {
        if (t < off) red[t] += red[t + off];
        __syncthreads();
    }
    if (t == 0) partial[blockIdx.x] = red[0];
}

// ---------------------------------------------------------------------------
// 5) Final reduction -> dictionary_loss == commitment_loss == MSE.
// ---------------------------------------------------------------------------
__global__ void k_reduce(const float* __restrict__ partial, float* __restrict__ outLoss) {
    __shared__ float red[256];
    const int t = threadIdx.x;
    float s = 0.f;
    for (int i = t; i < 4096; i += 256) s += partial[i];
    red[t] = s;
    __syncthreads();
    for (int off = 128; off > 0; off >>= 1) {
        if (t < off) red[t] += red[t + off];
        __syncthreads();
    }
    if (t == 0) {
        float mean = red[0] * (1.0f / 8388608.0f);
        outLoss[0] = mean;  // dictionary_loss
        outLoss[1] = mean;  // commitment_loss
    }
}

extern "C" void kernel_launch(void* const* d_in, const int* in_sizes, int n_in,
                              void* d_out, int out_size, void* d_ws, size_t ws_size,
                              hipStream_t stream) {
    const float* X = (const float*)d_in[0];  // [32768, 256]
    const float* V = (const float*)d_in[1];  // [256, 4096]
    float* out = (float*)d_out;

    // workspace carve (floats): VT | vnorm | idx(int) | partials
    float* ws    = (float*)d_ws;
    float* VT    = ws;                        // 4096*256 = 1048576 floats
    float* vnorm = ws + 1048576;              // 4096 floats
    int*   idx   = (int*)(ws + 1052672);      // 32768 ints
    float* part  = ws + 1085440;              // 4096 floats

    float* outQ    = out;                     // 8388608 floats
    float* outLoss = out + 8388608;           // 2 floats
    float* outIdxF = out + 8388610;           // 32768 floats (indices as float)

    k_transpose<<<dim3(NUM_K / 32, EMB_D / 32), dim3(32, 8), 0, stream>>>(V, VT);
    k_vnorm<<<NUM_K / 256, 256, 0, stream>>>(V, vnorm);
    k_argmin<<<NROWS / M_TILE, 256, 0, stream>>>(X, V, vnorm, idx, outIdxF);
    k_gather<<<NROWS / 8, 256, 0, stream>>>(X, VT, idx, outQ, part);
    k_reduce<<<1, 256, 0, stream>>>(part, outLoss);
}